// SimplifiedMultiHeadAttention_57878979281444
// MI455X (gfx1250) — compile-verified
//
#include <hip/hip_runtime.h>
#include <cstddef>

#define NB     8
#define SEQ    1024
#define EMBED  64
#define HEADS  8
#define CATW   (HEADS * EMBED)   // 512

typedef __attribute__((ext_vector_type(2))) float v2f;
typedef __attribute__((ext_vector_type(8))) float v8f;

// 128-bit transfer unit types for the async global<->LDS builtins:
// parameter type is int __attribute__((vector_size(16))) in as1/as3.
typedef int i32x4 __attribute__((vector_size(16)));
typedef __attribute__((address_space(1))) i32x4 g_i32x4;   // global
typedef __attribute__((address_space(3))) i32x4 l_i32x4;   // LDS

// ---------------------------------------------------------------------------
// V_WMMA_F32_16X16X4_F32 wrapper (codegen-confirmed).
// ---------------------------------------------------------------------------
__device__ __forceinline__ v8f wmma4(v2f a, v2f b, v8f c) {
  return __builtin_amdgcn_wmma_f32_16x16x4_f32(
      /*neg_a=*/false, a, /*neg_b=*/false, b,
      /*c_mod=*/(short)0, c, /*reuse_a=*/false, /*reuse_b=*/false);
}

// A fragment (16x4, MxK): lane l holds M = l%16, K = k0 + 2*(l/16) + {0,1}
__device__ __forceinline__ v2f load_a_frag(const float* __restrict__ A, int ld,
                                           int m0, int k0, int lane) {
  const float* p = A + (size_t)(m0 + (lane & 15)) * ld + (k0 + 2 * (lane >> 4));
  v2f a; a.x = p[0]; a.y = p[1]; return a;
}

// B fragment where B = M2^T (B[k][n] = M2[n][k]); lane holds N, VGPRs hold K.
__device__ __forceinline__ v2f load_bT_frag(const float* __restrict__ M2, int ld,
                                            int n0, int k0, int lane) {
  const float* p = M2 + (size_t)(n0 + (lane & 15)) * ld + (k0 + 2 * (lane >> 4));
  v2f b; b.x = p[0]; b.y = p[1]; return b;
}

// B fragment from a row-major B[k][n] directly.
__device__ __forceinline__ v2f load_b_frag(const float* __restrict__ B, int ld,
                                           int k0, int n0, int lane) {
  int n = n0 + (lane & 15);
  int k = k0 + 2 * (lane >> 4);
  v2f b; b.x = B[(size_t)k * ld + n]; b.y = B[(size_t)(k + 1) * ld + n]; return b;
}

// D store: VGPR r -> row m0 + r + 8*(lane/16); lane -> col n0 + lane%16.
__device__ __forceinline__ void store_d_frag(float* __restrict__ O, int ld,
                                             int m0, int n0, int lane, v8f d,
                                             float scale) {
  int col   = n0 + (lane & 15);
  int rbase = m0 + 8 * (lane >> 4);
#pragma unroll
  for (int r = 0; r < 8; ++r) O[(size_t)(rbase + r) * ld + col] = d[r] * scale;
}

// ---------------------------------------------------------------------------
// CDNA5 async global<->LDS block transfers (ASYNCcnt-tracked), with a
// synchronous fallback so the kernel compiles on any toolchain.
// ---------------------------------------------------------------------------
__device__ __forceinline__ void async_g2l_16B(const float* g, float* l) {
#if __has_builtin(__builtin_amdgcn_global_load_async_to_lds_b128)
  __builtin_amdgcn_global_load_async_to_lds_b128(
      (g_i32x4*)g, (l_i32x4*)l, 0, 0);
#else
  *(float4*)l = *(const float4*)g;
#endif
}

__device__ __forceinline__ void async_l2g_16B(float* g, const float* l) {
#if __has_builtin(__builtin_amdgcn_global_store_async_from_lds_b128)
  __builtin_amdgcn_global_store_async_from_lds_b128(
      (g_i32x4*)g, (l_i32x4*)l, 0, 0);
#else
  *(float4*)g = *(const float4*)l;
#endif
}

__device__ __forceinline__ void wait_async0() {
#if __has_builtin(__builtin_amdgcn_s_wait_asynccnt)
  __builtin_amdgcn_s_wait_asynccnt(0);
#elif __has_builtin(__builtin_amdgcn_global_load_async_to_lds_b128)
  asm volatile("s_wait_asynccnt 0" ::: "memory");
#endif
}

// ---------------------------------------------------------------------------
// Kernel 1: per-head input projections  Y[n,h] = X[n] @ W[h]^T   (S x E)
// grid (N*H, S/128), 256 threads (8 waves); wave owns one 16-row stripe.
// ---------------------------------------------------------------------------
__device__ __forceinline__ void proj_stripe(const float* __restrict__ X,
                                            const float* __restrict__ W,
                                            float* __restrict__ Y,
                                            int m0, int lane) {
  v8f z = {};
  v8f acc[4] = {z, z, z, z};
#pragma unroll
  for (int k0 = 0; k0 < EMBED; k0 += 4) {
    v2f a = load_a_frag(X, EMBED, m0, k0, lane);
#pragma unroll
    for (int i = 0; i < 4; ++i) {
      v2f b = load_bT_frag(W, EMBED, i * 16, k0, lane);   // B = W^T
      acc[i] = wmma4(a, b, acc[i]);
    }
  }
#pragma unroll
  for (int i = 0; i < 4; ++i) store_d_frag(Y, EMBED, m0, i * 16, lane, acc[i], 1.0f);
}

__global__ void __launch_bounds__(256) proj_all_kernel(
    const float* __restrict__ V, const float* __restrict__ K,
    const float* __restrict__ Q, const float* __restrict__ Wv,
    const float* __restrict__ Wk, const float* __restrict__ Wq,
    float* __restrict__ Vp, float* __restrict__ Kp, float* __restrict__ Qp) {
  const int nh = blockIdx.x;
  const int n = nh / HEADS, h = nh % HEADS;
  const int lane = threadIdx.x & 31, wave = threadIdx.x >> 5;
  const int m0 = blockIdx.y * 128 + wave * 16;

  const size_t xoff = (size_t)n * SEQ * EMBED;
  const size_t woff = (size_t)h * EMBED * EMBED;
  const size_t yoff = (size_t)nh * SEQ * EMBED;

  proj_stripe(Q + xoff, Wq + woff, Qp + yoff, m0, lane);
  proj_stripe(K + xoff, Wk + woff, Kp + yoff, m0, lane);
  proj_stripe(V + xoff, Wv + woff, Vp + yoff, m0, lane);
}

// ---------------------------------------------------------------------------
// Kernel 2: energy -> softmax -> async attn writeback -> head_out = P @ V
// grid (N*H, S/32), 256 threads.
// LDS: 128 KB prob tile + 16 KB K/V staging (~145 KB => 2 WGs / WGP).
// ---------------------------------------------------------------------------
#define CH 64   // K/V chunk rows staged per step (64 x 64 fp32 = 16 KB)

__global__ void __launch_bounds__(256) attn_kernel(
    const float* __restrict__ Qp, const float* __restrict__ Kp,
    const float* __restrict__ Vp, float* __restrict__ attn,
    float* __restrict__ Cat) {
  __shared__ float sE[32 * SEQ];      // 128 KB energy / prob tile
  __shared__ float sKV[CH * EMBED];   // 16 KB K/V staging buffer
  __shared__ float sPart[256];
  __shared__ float sMax[32];
  __shared__ float sSum[32];

  const int nh = blockIdx.x;
  const int q0 = blockIdx.y * 32;
  const int lane = threadIdx.x & 31, wave = threadIdx.x >> 5;
  const int tid = threadIdx.x;

  const float* Qh = Qp + (size_t)nh * SEQ * EMBED;
  const float* Kh = Kp + (size_t)nh * SEQ * EMBED;
  const float* Vh = Vp + (size_t)nh * SEQ * EMBED;

  const int mt = wave & 1;        // row-tile (0..1) owned by this wave
  const int ct = wave >> 1;       // col-tile (0..3) within a chunk

  // Q A-fragments are K-chunk-invariant: load them ONCE into registers.
  v2f aq[EMBED / 4];
#pragma unroll
  for (int k0 = 0; k0 < EMBED; k0 += 4)
    aq[k0 / 4] = load_a_frag(Qh, EMBED, q0 + mt * 16, k0, lane);

  // ---- energy = (Q @ K^T)/8, K streamed through LDS in 16 chunks of 64 rows
  for (int kc = 0; kc < SEQ / CH; ++kc) {
    const float* src = Kh + (size_t)kc * CH * EMBED;    // contiguous 16 KB
#pragma unroll
    for (int i = tid; i < CH * EMBED / 4; i += 256)
      async_g2l_16B(src + i * 4, sKV + i * 4);
    wait_async0();
    __syncthreads();

    v8f acc = {};
#pragma unroll
    for (int k0 = 0; k0 < EMBED; k0 += 4) {
      v2f b = load_bT_frag(sKV, EMBED, ct * 16, k0, lane);   // B = Kchunk^T
      acc = wmma4(aq[k0 / 4], b, acc);
    }
    store_d_frag(sE, SEQ, mt * 16, kc * CH + ct * 16, lane, acc, 0.125f);
    __syncthreads();   // chunk consumed; safe to overwrite sKV
  }

  // ---- row softmax over 1024 keys; 8 threads per row
  const int r = tid >> 3;
  const int sub = tid & 7;
  float* row = sE + r * SEQ;

  float m = -3.402823466e+38f;
  for (int c = sub; c < SEQ; c += 8) m = fmaxf(m, row[c]);
  sPart[tid] = m;
  __syncthreads();
  if (sub == 0) {
    float mm = sPart[tid];
#pragma unroll
    for (int j = 1; j < 8; ++j) mm = fmaxf(mm, sPart[tid + j]);
    sMax[r] = mm;
  }
  __syncthreads();

  const float rm = sMax[r];
  float s = 0.0f;
  for (int c = sub; c < SEQ; c += 8) {
    float e = __expf(row[c] - rm);
    row[c] = e;
    s += e;
  }
  sPart[tid] = s;
  __syncthreads();
  if (sub == 0) {
    float ss = sPart[tid];
#pragma unroll
    for (int j = 1; j < 8; ++j) ss += sPart[tid + j];
    sSum[r] = ss;
  }
  __syncthreads();

  const float inv = 1.0f / sSum[r];
  for (int c = sub; c < SEQ; c += 8) row[c] *= inv;
  __syncthreads();

  // ---- attn writeback: the 32 rows are CONTIGUOUS in memory and match sE's
  //      layout exactly -> pure LDS->HBM block DMA (this is the HBM floor).
  {
    float* dst = attn + ((size_t)nh * SEQ + q0) * SEQ;   // 32*SEQ floats
    for (int i = tid; i < 32 * SEQ / 4; i += 256)
      async_l2g_16B(dst + i * 4, sE + i * 4);
    // No wait needed before P@V: these stores only READ sE, which is never
    // modified again; kernel-end implicitly drains all counters.
  }

  // ---- head_out = P @ V : 8 tiles (2x4), one per wave; V staged via LDS
  {
    const int nt = wave >> 1;     // output col-tile (0..3)
    v8f acc = {};
    for (int vc = 0; vc < SEQ / CH; ++vc) {
      const float* src = Vh + (size_t)vc * CH * EMBED;
#pragma unroll
      for (int i = tid; i < CH * EMBED / 4; i += 256)
        async_g2l_16B(src + i * 4, sKV + i * 4);
      wait_async0();
      __syncthreads();
#pragma unroll
      for (int k0 = 0; k0 < CH; k0 += 4) {
        v2f a = load_a_frag(sE, SEQ, mt * 16, vc * CH + k0, lane);  // P (LDS)
        v2f b = load_b_frag(sKV, EMBED, k0, nt * 16, lane);         // V (LDS)
        acc = wmma4(a, b, acc);
      }
      __syncthreads();   // chunk consumed; safe to overwrite sKV
    }
    const int n = nh / HEADS, h = nh % HEADS;
    float* Cn = Cat + (size_t)n * SEQ * CATW;   // [S, H*E] for batch n
    store_d_frag(Cn, CATW, q0 + mt * 16, h * EMBED + nt * 16, lane, acc, 1.0f);
  }
}

// ---------------------------------------------------------------------------
// Kernel 3: out = Cat @ Wout^T + bout   (M=8192, K=512, N=64)
// ---------------------------------------------------------------------------
__global__ void __launch_bounds__(256) outproj_kernel(
    const float* __restrict__ Cat, const float* __restrict__ Wout,
    const float* __restrict__ bout, float* __restrict__ out) {
  const int lane = threadIdx.x & 31, wave = threadIdx.x >> 5;
  const int m0 = blockIdx.x * 128 + wave * 16;

  v8f z = {};
  v8f acc[4] = {z, z, z, z};
  for (int k0 = 0; k0 < CATW; k0 += 4) {
    v2f a = load_a_frag(Cat, CATW, m0, k0, lane);
#pragma unroll
    for (int i = 0; i < 4; ++i) {
      v2f b = load_bT_frag(Wout, CATW, i * 16, k0, lane);   // B = Wout^T
      acc[i] = wmma4(a, b, acc[i]);
    }
  }
#pragma unroll
  for (int i = 0; i < 4; ++i) {
    float bb = bout[i * 16 + (lane & 15)];
    acc[i] += bb;
    store_d_frag(out, EMBED, m0, i * 16, lane, acc[i], 1.0f);
  }
}

// ---------------------------------------------------------------------------
extern "C" void kernel_launch(void* const* d_in, const int* in_sizes, int n_in,
                              void* d_out, int out_size, void* d_ws, size_t ws_size,
                              hipStream_t stream) {
  (void)in_sizes; (void)n_in; (void)out_size; (void)ws_size;

  const float* V    = (const float*)d_in[0];
  const float* K    = (const float*)d_in[1];
  const float* Q    = (const float*)d_in[2];
  const float* Wv   = (const float*)d_in[3];
  const float* Wk   = (const float*)d_in[4];
  const float* Wq   = (const float*)d_in[5];
  const float* Wout = (const float*)d_in[6];
  const float* bout = (const float*)d_in[7];

  float* out  = (float*)d_out;                              // [N,S,E]
  float* attn = out + (size_t)NB * SEQ * EMBED;             // [N,H,S,S]

  // workspace: Qp | Kp | Vp | Cat  (16 MB each, 64 MB total)
  const size_t PROJ = (size_t)NB * HEADS * SEQ * EMBED;     // 4M floats
  float* ws = (float*)d_ws;
  float* Qp  = ws;
  float* Kp  = ws + PROJ;
  float* Vp  = ws + 2 * PROJ;
  float* Cat = ws + 3 * PROJ;

  proj_all_kernel<<<dim3(NB * HEADS, SEQ / 128), 256, 0, stream>>>(
      V, K, Q, Wv, Wk, Wq, Vp, Kp, Qp);
  attn_kernel<<<dim3(NB * HEADS, SEQ / 32), 256, 0, stream>>>(
      Qp, Kp, Vp, attn, Cat);
  outproj_kernel<<<dim3((NB * SEQ) / 128), 256, 0, stream>>>(
      Cat, Wout, bout, out);
}